// PassengerStateTransition_53807350284622
// MI455X (gfx1250) — compile-verified
//
#include <hip/hip_runtime.h>
#include <stdint.h>

// ---------------------------------------------------------------------------
// PassengerStateTransition for MI455X (gfx1250)
//   Kernel 1: bulk copy of the 176MB passenger table (HBM-bound). Uses the
//             gfx1250 async global->LDS->global B128 path; full blocks issue
//             4 loads + 4 stores back-to-back using literal IOFFSETs (the
//             offset is applied to both LDS and global addresses per ISA),
//             with a single ASYNCcnt wait in between.
//   Kernel 2: one workgroup per env: duplicate resolution in LDS (u64 atomicMin
//             keyed on (dist_bits<<32)|agent), then scatter-patch cols 6/7/9/10
//             of modified rows directly in d_out (stream-ordered after copy).
// ---------------------------------------------------------------------------

#define INFF __builtin_inff()

#if __has_builtin(__builtin_amdgcn_global_load_async_to_lds_b128) && \
    __has_builtin(__builtin_amdgcn_global_store_async_from_lds_b128)
#define PST_ASYNC_LDS 1
#else
#define PST_ASYNC_LDS 0
#endif

// Builtin signature (probe-confirmed by hipcc diagnostics): pointers to
// 4 x int vectors, global (AS1) source/dest and LDS (AS3) stage; the offset
// argument must be a literal constant integer.
typedef int pst_v4i __attribute__((ext_vector_type(4)));
typedef __attribute__((address_space(1))) pst_v4i pst_gv4i;
typedef __attribute__((address_space(3))) pst_v4i pst_lv4i;

__device__ __forceinline__ void pst_wait_async0() {
#if __has_builtin(__builtin_amdgcn_s_wait_asynccnt)
  __builtin_amdgcn_s_wait_asynccnt(0);
#else
  asm volatile("s_wait_asynccnt 0x0" ::: "memory");
#endif
}

// ------------------------- bulk table copy ---------------------------------
// Each 256-thread block moves 1024 float4s (16KB) through LDS.
#define PST_COPY_VEC_PER_BLOCK 1024

#define PST_ASYNC_LOAD(g, l, off) \
  __builtin_amdgcn_global_load_async_to_lds_b128((pst_gv4i*)(g), (pst_lv4i*)(l), (off), 0)
#define PST_ASYNC_STORE(g, l, off) \
  __builtin_amdgcn_global_store_async_from_lds_b128((pst_gv4i*)(g), (pst_lv4i*)(l), (off), 0)

__global__ __launch_bounds__(256) void pst_copy(const float* __restrict__ in,
                                                float* __restrict__ out,
                                                long long n) {
  const long long n4   = n >> 2;
  const long long base = (long long)blockIdx.x * PST_COPY_VEC_PER_BLOCK;
  const int       tid  = threadIdx.x;
#if PST_ASYNC_LDS
  __shared__ float4 stage[PST_COPY_VEC_PER_BLOCK];
  const float4* in4  = reinterpret_cast<const float4*>(in);
  float4*       out4 = reinterpret_cast<float4*>(out);
  if (base + PST_COPY_VEC_PER_BLOCK <= n4) {
    // Fast path: full block, no guards. IOFFSET advances both the LDS stage
    // address and the global address,
    // so one base address pair serves all 4 transfers.
    const float4* gsrc = in4 + (base + tid);
    float4*       gdst = out4 + (base + tid);
    float4*       lptr = &stage[tid];
    PST_ASYNC_LOAD(gsrc, lptr, 0);
    PST_ASYNC_LOAD(gsrc, lptr, 4096);
    PST_ASYNC_LOAD(gsrc, lptr, 8192);
    PST_ASYNC_LOAD(gsrc, lptr, 12288);
    pst_wait_async0();
    PST_ASYNC_STORE(gdst, lptr, 0);
    PST_ASYNC_STORE(gdst, lptr, 4096);
    PST_ASYNC_STORE(gdst, lptr, 8192);
    PST_ASYNC_STORE(gdst, lptr, 12288);
  } else {
#pragma unroll
    for (int k = 0; k < 4; ++k) {
      const long long i = base + (long long)k * 256 + tid;
      if (i < n4) {
        PST_ASYNC_LOAD(in4 + i, &stage[k * 256 + tid], 0);
      }
    }
    pst_wait_async0();
#pragma unroll
    for (int k = 0; k < 4; ++k) {
      const long long i = base + (long long)k * 256 + tid;
      if (i < n4) {
        PST_ASYNC_STORE(out4 + i, &stage[k * 256 + tid], 0);
      }
    }
  }
  // No trailing ASYNCcnt wait needed: S_ENDPGM performs an implicit wait-idle.
#else
#pragma unroll
  for (int k = 0; k < 4; ++k) {
    const long long i = base + (long long)k * 256 + tid;
    if (i < n4) {
      reinterpret_cast<float4*>(out)[i] =
          reinterpret_cast<const float4*>(in)[i];
    }
  }
#endif
  const long long rem = n & 3;
  if (rem && base == 0 && tid == 0) {
    for (long long t = n - rem; t < n; ++t) out[t] = in[t];
  }
}

// ------------------- per-env resolve + scatter patch -----------------------
#define PST_MAXK 512

__global__ __launch_bounds__(256) void pst_resolve(
    const float* __restrict__ passengers,
    const unsigned char* __restrict__ accepts,
    const unsigned char* __restrict__ picks,
    const int* __restrict__ targets,
    const float* __restrict__ vectors,
    const int* __restrict__ timesteps,
    float* __restrict__ out,
    int A, int K) {
  __shared__ unsigned cnt[PST_MAXK];     // accept-target histogram
  __shared__ unsigned state[PST_MAXK];   // bit0=accepted, bit1=picked, bits2..=agent
  __shared__ unsigned long long minKey;  // (dist_bits<<32)|agent over dup agents

  const int e = blockIdx.x;
  const int tid = threadIdx.x;

  for (int i = tid; i < K; i += blockDim.x) { cnt[i] = 0u; state[i] = 0u; }
  if (tid == 0) minKey = ~0ull;
  __syncthreads();

  int   accept_lt = -1;
  int   lt        = -1;
  float dist      = INFF;
  bool  dup       = false;

  if (tid < A) {
    const long long idx = (long long)e * A + tid;
    const float v0 = vectors[idx * 4 + 0];
    const float v1 = vectors[idx * 4 + 1];
    const float v2 = vectors[idx * 4 + 2];
    const float v3 = vectors[idx * 4 + 3];
    const bool invalid =
        (v0 == -100.0f) & (v1 == -100.0f) & (v2 == -100.0f) & (v3 == -100.0f);
    const float dx = v0 - v2, dy = v1 - v3;
    dist = invalid ? INFF : sqrtf(dx * dx + dy * dy);
    lt = targets[idx] - e * K;  // env-partitioned -> local index in [0,K)
    if (accepts[idx]) accept_lt = lt;
    if (accept_lt >= 0) atomicAdd(&cnt[accept_lt], 1u);
  }
  __syncthreads();

  // dup = valid accept whose target is shared by >=2 valid accepts in this env
  if (accept_lt >= 0 && cnt[accept_lt] >= 2u) {
    dup = true;
    const unsigned long long key =
        ((unsigned long long)__float_as_uint(dist) << 32) | (unsigned)tid;
    atomicMin(&minKey, key);  // ds_min_u64 on LDS
  }
  __syncthreads();

  if (dup) {
    const unsigned long long mk = minKey;
    // If every dup agent has inf distance, the reference argmin over the
    // all-inf masked row is index 0.
    const int keeper =
        ((unsigned)(mk >> 32) == 0x7f800000u) ? 0 : (int)(unsigned)mk;
    if (tid != keeper) accept_lt = -1;
  }

  // accept scatter (at most one agent per passenger after dedup)
  if (accept_lt >= 0) atomicOr(&state[accept_lt], 1u | ((unsigned)tid << 2));
  // pick scatter: only zero-distance picks succeed (invalid agents have inf)
  if (tid < A) {
    const long long idx = (long long)e * A + tid;
    if (picks[idx] && dist < 1e-6f) atomicOr(&state[lt], 2u);
  }
  __syncthreads();

  // Patch modified rows. One thread per passenger => accept-then-pick ordering
  // within the thread reproduces the reference's column-6 override (2.0 wins).
  for (int lp = tid; lp < K; lp += blockDim.x) {
    const unsigned s = state[lp];
    if (!s) continue;
    const long long r = (long long)e * K + lp;
    float* rowo = out + r * 11;
    const int env = (int)passengers[r * 11 + 0];
    const float ts = (float)timesteps[env];
    if (s & 1u) {
      rowo[6] = 1.0f;
      rowo[9] = ts;
      rowo[7] = (float)((s >> 2) & 0x3ffu);
    }
    if (s & 2u) {
      rowo[6] = 2.0f;
      rowo[10] = ts;
    }
  }
}

// ------------------------------- launcher ----------------------------------
extern "C" void kernel_launch(void* const* d_in, const int* in_sizes, int n_in,
                              void* d_out, int out_size, void* d_ws, size_t ws_size,
                              hipStream_t stream) {
  (void)n_in; (void)d_ws; (void)ws_size;
  const float*         passengers = (const float*)d_in[0];
  const unsigned char* accepts    = (const unsigned char*)d_in[1];  // jax bool = 1 byte
  const unsigned char* picks      = (const unsigned char*)d_in[2];
  const int*           targets    = (const int*)d_in[3];
  const float*         vectors    = (const float*)d_in[4];
  const int*           timesteps  = (const int*)d_in[5];
  float*               out        = (float*)d_out;

  const long long P = (long long)in_sizes[0] / 11;
  const int       E = in_sizes[5];
  const int       A = in_sizes[1] / E;
  const int       K = (int)(P / E);

  const long long n  = (long long)out_size;     // P*11 floats
  const long long n4 = n >> 2;
  const long long cg = (n4 + PST_COPY_VEC_PER_BLOCK - 1) / PST_COPY_VEC_PER_BLOCK;

  pst_copy<<<dim3((unsigned)cg), dim3(256), 0, stream>>>(passengers, out, n);
  pst_resolve<<<dim3((unsigned)E), dim3(256), 0, stream>>>(
      passengers, accepts, picks, targets, vectors, timesteps, out, A, K);
}